// DiffJPEGCoding_7705171329349
// MI455X (gfx1250) — compile-verified
//
#include <hip/hip_runtime.h>
#include <math.h>

typedef float v2f __attribute__((ext_vector_type(2)));
typedef float v8f __attribute__((ext_vector_type(8)));

// C[x][u] = cos((2x+1)*u*pi/16)  (DCT-II basis)
__constant__ float CTAB[64] = {
  1.0f,  0.98078528f,  0.92387953f,  0.83146961f,  0.70710678f,  0.55557023f,  0.38268343f,  0.19509032f,
  1.0f,  0.83146961f,  0.38268343f, -0.19509032f, -0.70710678f, -0.98078528f, -0.92387953f, -0.55557023f,
  1.0f,  0.55557023f, -0.38268343f, -0.98078528f, -0.70710678f,  0.19509032f,  0.92387953f,  0.83146961f,
  1.0f,  0.19509032f, -0.92387953f, -0.55557023f,  0.70710678f,  0.83146961f, -0.38268343f, -0.98078528f,
  1.0f, -0.19509032f, -0.92387953f,  0.55557023f,  0.70710678f, -0.83146961f, -0.38268343f,  0.98078528f,
  1.0f, -0.55557023f, -0.38268343f,  0.98078528f, -0.70710678f, -0.19509032f,  0.92387953f, -0.83146961f,
  1.0f, -0.83146961f,  0.38268343f,  0.19509032f, -0.70710678f,  0.98078528f, -0.92387953f,  0.55557023f,
  1.0f, -0.98078528f,  0.92387953f, -0.83146961f,  0.70710678f, -0.55557023f,  0.38268343f, -0.19509032f
};

__constant__ float QTY[64] = {
  16,11,10,16,24,40,51,61, 12,12,14,19,26,58,60,55,
  14,13,16,24,40,57,69,56, 14,17,22,29,51,87,80,62,
  18,22,37,56,68,109,103,77, 24,35,55,64,81,104,113,92,
  49,64,78,87,103,121,120,101, 72,92,95,98,112,100,103,99
};
__constant__ float QTC[64] = {
  17,18,24,47,99,99,99,99, 18,21,26,66,99,99,99,99,
  24,26,56,99,99,99,99,99, 47,66,99,99,99,99,99,99,
  99,99,99,99,99,99,99,99, 99,99,99,99,99,99,99,99,
  99,99,99,99,99,99,99,99, 99,99,99,99,99,99,99,99
};

__device__ __forceinline__ float poly_round_f(float x) {
  float r = rintf(x);            // v_rndne_f32: matches jnp.round (half-even)
  float d = x - r;
  return r + d * d * d;
}

// qt_scaled with q == FACTOR == 50: s = poly_floor(200-2*50) = 99.875 (const)
// poly_floor((qt*99.875+50)/100) == poly_round(qt*0.99875): no division needed.
__device__ __forceinline__ float qscaled(float qt) {
  return poly_round_f(qt * 0.99875f);  // diff_clip(1,255) identity in-range
}

// element (r,c) of block-diagonal diag(C,C); zero off 8x8 diagonal blocks
__device__ __forceinline__ float maskedC(int r, int c) {
  float v = CTAB[(r & 7) * 8 + (c & 7)];
  return ((r ^ c) & 8) ? 0.0f : v;
}

__device__ __forceinline__ v8f wmma4(v2f a, v2f b, v8f c) {
  // V_WMMA_F32_16X16X4_F32: D = A(16x4) x B(4x16) + C(16x16), f32 throughout
  return __builtin_amdgcn_wmma_f32_16x16x4_f32(false, a, false, b, (short)0, c,
                                               false, false);
}

// store 16x16 f32 D-layout accumulator into LDS (row stride 17)
__device__ __forceinline__ void storeD(float* w, int lane, v8f acc) {
  int n = lane & 15;
  int hi = (lane >> 4) << 3;           // lanes 16-31 hold rows 8..15
#pragma unroll
  for (int i = 0; i < 8; ++i) w[(i + hi) * 17 + n] = acc[i];
}

// acc = const(A, registers) x data(LDS, as B); 4 chained K=4 WMMAs
__device__ __forceinline__ v8f mulConstA(const float* w, int lane, const v2f* ca) {
  v8f acc = {0, 0, 0, 0, 0, 0, 0, 0};
  int n = lane & 15;
  int koff = (lane >> 4) << 1;         // lanes 16-31 carry K+2,K+3
#pragma unroll
  for (int j = 0; j < 4; ++j) {
    int kb = 4 * j + koff;
    v2f b;
    b.x = w[kb * 17 + n];
    b.y = w[(kb + 1) * 17 + n];
    acc = wmma4(ca[j], b, acc);
  }
  return acc;
}

// acc = data(LDS, as A) x const(B, registers)
__device__ __forceinline__ v8f mulConstB(const float* w, int lane, const v2f* cb) {
  v8f acc = {0, 0, 0, 0, 0, 0, 0, 0};
  int m = lane & 15;
  int koff = (lane >> 4) << 1;
#pragma unroll
  for (int j = 0; j < 4; ++j) {
    int kb = 4 * j + koff;
    v2f a;
    a.x = w[m * 17 + kb];
    a.y = w[m * 17 + kb + 1];
    acc = wmma4(a, cb[j], acc);
  }
  return acc;
}

// Full DCT -> quantize -> dequantize -> IDCT on one 16x16 LDS buffer
// (4 independent 8x8 blocks via block-diagonal basis). In-place.
// cKM[j] = C[k][m]-type operand (fwd both sides), cMK[j] = C[m][k]-type (inv).
__device__ __forceinline__ void jpeg8x8(float* w, int lane, const float* qtab,
                                        const v2f* cKM, const v2f* cMK) {
  int n = lane & 15;
  // hoist lane-divergent quant loads so they overlap the WMMA chains
  float k1[8], k2[8];
#pragma unroll
  for (int i = 0; i < 8; ++i) {        // m&7 == i regardless of lane half
    float qt = qscaled(qtab[i * 8 + (n & 7)]);
    float am = (i == 0) ? 0.70710678f : 1.0f;
    float an = ((n & 7) == 0) ? 0.70710678f : 1.0f;
    float k = 0.25f * am * an;         // 0.25 * alpha2
    k1[i] = k * __builtin_amdgcn_rcpf(qt);  // v_rcp_f32: no div expansion
    k2[i] = k * qt;
  }
  v8f u = mulConstA(w, lane, cKM);     // diag(C^T,C^T) * P
  storeD(w, lane, u);
  __syncthreads();                     // single-wave WG: barrier is S_NOP
  v8f t = mulConstB(w, lane, cKM);     // ... * diag(C,C)   => per-block C^T P C
  v8f wq;
#pragma unroll
  for (int i = 0; i < 8; ++i)
    wq[i] = poly_round_f(t[i] * k1[i]) * k2[i];  // quantize -> dequantize
  storeD(w, lane, wq);
  __syncthreads();
  v8f v = mulConstA(w, lane, cMK);     // diag(C,C) * W
  storeD(w, lane, v);
  __syncthreads();
  v8f x = mulConstB(w, lane, cMK);     // ... * diag(C^T,C^T) => per-block C W C^T
  storeD(w, lane, x);
  __syncthreads();
}

__device__ __forceinline__ float dclip255(float x) {
  float out = x;
  out = (x > 255.0f) ? (-0.02f * (expf(255.0f - x) - 1.0f) + 255.0f) : out;
  out = (x < 0.0f) ? (0.02f * (expf(x) - 1.0f)) : out;
  return out;
}

// One wave32 workgroup per 16x16 pixel tile. Single-wave WGs get no barrier
// resource (ISA 3.1): s_barrier_* execute as S_NOP, so the LDS ping-pong
// stages cost only the s_wait_dscnt the compiler inserts anyway.
__global__ __launch_bounds__(32) void jpeg_kernel(const float* __restrict__ in,
                                                  float* __restrict__ out) {
  __shared__ float smem[4][272];       // y, cb, cr, packed-chroma (stride 17)
  const int lane = threadIdx.x & 31;
  const int tile = blockIdx.x;         // 32768 tiles total
  const int b = tile >> 10;
  const int ty = (tile >> 5) & 31;
  const int tx = tile & 31;
  float* wy = smem[0];
  float* wc = smem[1];
  float* wt = smem[2];
  float* wp = smem[3];

  // ---- per-lane constant WMMA operands (computed once, 16 VGPRs) ----
  v2f cKM[4], cMK[4];
  {
    int m = lane & 15;
    int koff = (lane >> 4) << 1;
#pragma unroll
    for (int j = 0; j < 4; ++j) {
      int kb = 4 * j + koff;
      cKM[j].x = maskedC(kb, m);       // C[k][m]  (C^T as A / C as B)
      cKM[j].y = maskedC(kb + 1, m);
      cMK[j].x = maskedC(m, kb);       // C[m][k]  (C as A / C^T as B)
      cMK[j].y = maskedC(m, kb + 1);
    }
  }

  const long long plane = 512LL * 512LL;
  const float* base = in + (long long)b * 3 * plane;

  // ---- load 16x16 RGB tile (float4, coalesced), RGB -> YCbCr (centered) ----
#pragma unroll
  for (int p = 0; p < 2; ++p) {
    int row = p * 8 + (lane >> 2);
    int col = (lane & 3) * 4;
    long long off = (long long)(ty * 16 + row) * 512 + tx * 16 + col;
    float4 R = *(const float4*)(base + 0 * plane + off);
    float4 G = *(const float4*)(base + 1 * plane + off);
    float4 B = *(const float4*)(base + 2 * plane + off);
    float rv[4] = {R.x, R.y, R.z, R.w};
    float gv[4] = {G.x, G.y, G.z, G.w};
    float bv[4] = {B.x, B.y, B.z, B.w};
#pragma unroll
    for (int i = 0; i < 4; ++i) {
      float r = (fminf(fmaxf(rv[i], -1.0f), 1.0f) + 1.0f) * 127.5f;
      float g = (fminf(fmaxf(gv[i], -1.0f), 1.0f) + 1.0f) * 127.5f;
      float bb = (fminf(fmaxf(bv[i], -1.0f), 1.0f) + 1.0f) * 127.5f;
      float y = 0.299f * r + 0.587f * g + 0.114f * bb - 128.0f;
      float cb = -0.168736f * r - 0.331264f * g + 0.5f * bb;  // cb-128
      float cr = 0.5f * r - 0.418688f * g - 0.081312f * bb;   // cr-128
      wy[row * 17 + col + i] = y;
      wc[row * 17 + col + i] = cb;
      wt[row * 17 + col + i] = cr;
    }
  }
  // zero packed-chroma buffer (off-diagonal blocks must be 0)
  for (int i = lane; i < 272; i += 32) wp[i] = 0.0f;
  __syncthreads();
  // ---- 2x2 avgpool chroma into block-diagonal packed 16x16: [Cb 0; 0 Cr] ----
  {
    int u = lane >> 2;
    int v0 = (lane & 3) * 2;
#pragma unroll
    for (int dv = 0; dv < 2; ++dv) {
      int v = v0 + dv;
      int r0 = (2 * u) * 17 + 2 * v, r1 = (2 * u + 1) * 17 + 2 * v;
      float cb = 0.25f * (wc[r0] + wc[r0 + 1] + wc[r1] + wc[r1 + 1]);
      float cr = 0.25f * (wt[r0] + wt[r0 + 1] + wt[r1] + wt[r1 + 1]);
      wp[u * 17 + v] = cb;
      wp[(u + 8) * 17 + (v + 8)] = cr;
    }
  }
  __syncthreads();

  // ---- JPEG round-trip: 4 Y blocks + packed Cb/Cr, all via f32 WMMA ----
  jpeg8x8(wy, lane, QTY, cKM, cMK);   // 16 x v_wmma_f32_16x16x4_f32
  jpeg8x8(wp, lane, QTC, cKM, cMK);   // 16 x v_wmma_f32_16x16x4_f32

  // ---- upsample chroma, YCbCr -> RGB, soft clip, rescale, store ----
  float* obase = out + (long long)b * 3 * plane;
#pragma unroll
  for (int p = 0; p < 2; ++p) {
    int row = p * 8 + (lane >> 2);
    int col = (lane & 3) * 4;
    long long off = (long long)(ty * 16 + row) * 512 + tx * 16 + col;
    float ro[4], go[4], bo[4];
#pragma unroll
    for (int i = 0; i < 4; ++i) {
      int c = col + i;
      float y = wy[row * 17 + c] + 128.0f;
      float cb = wp[(row >> 1) * 17 + (c >> 1)];            // already centered
      float cr = wp[((row >> 1) + 8) * 17 + (c >> 1) + 8];
      float r = y + 1.402f * cr;
      float g = y - 0.344136f * cb - 0.714136f * cr;
      float bb = y + 1.772f * cb;
      r = dclip255(r) * (2.0f / 255.0f) - 1.0f;
      g = dclip255(g) * (2.0f / 255.0f) - 1.0f;
      bb = dclip255(bb) * (2.0f / 255.0f) - 1.0f;
      ro[i] = fminf(fmaxf(r, -1.0f), 1.0f);
      go[i] = fminf(fmaxf(g, -1.0f), 1.0f);
      bo[i] = fminf(fmaxf(bb, -1.0f), 1.0f);
    }
    *(float4*)(obase + 0 * plane + off) = make_float4(ro[0], ro[1], ro[2], ro[3]);
    *(float4*)(obase + 1 * plane + off) = make_float4(go[0], go[1], go[2], go[3]);
    *(float4*)(obase + 2 * plane + off) = make_float4(bo[0], bo[1], bo[2], bo[3]);
  }
}

extern "C" void kernel_launch(void* const* d_in, const int* in_sizes, int n_in,
                              void* d_out, int out_size, void* d_ws, size_t ws_size,
                              hipStream_t stream) {
  const float* img = (const float*)d_in[0];
  float* outp = (float*)d_out;
  // 32 images * (512/16)^2 tiles = 32768 tiles; one wave32 workgroup per tile
  dim3 grid(32768), block(32);
  hipLaunchKernelGGL(jpeg_kernel, grid, block, 0, stream, img, outp);
}